// ClientModel_59081570125008
// MI455X (gfx1250) — compile-verified
//
#include <hip/hip_runtime.h>
#include <hip/hip_bf16.h>

typedef float v2f __attribute__((ext_vector_type(2)));
typedef float v8f __attribute__((ext_vector_type(8)));

constexpr int kV = 10000, kH = 512, kL = 2, kB = 128, kS = 20;
constexpr int k4H = 4 * kH;
constexpr int kRows = kB * kS;          // 2560 tokens
constexpr int kRowTiles = kRows / 16;   // 160
constexpr int kNTiles = kV / 16;        // 625 column tiles of the vocab
constexpr int kWorkers = 64;            // projection column workers (chunks)

__device__ __forceinline__ v8f wmma_f32(v2f a, v2f b, v8f c) {
  // V_WMMA_F32_16X16X4_F32 : D = A(16x4) * B(4x16) + C(16x16), all fp32
  return __builtin_amdgcn_wmma_f32_16x16x4_f32(false, a, false, b, (short)0, c,
                                               false, false);
}

__device__ __forceinline__ float sigmoidf_(float x) {
  return 1.0f / (1.0f + __expf(-x));
}

__global__ void zero_kernel(float* __restrict__ p, int n) {
  int i = blockIdx.x * blockDim.x + threadIdx.x;
  if (i < n) p[i] = 0.0f;
}

// x[t][b][h] = embedding[features[b][t]][h]
__global__ void embed_kernel(const int* __restrict__ features,
                             const float* __restrict__ emb,
                             float* __restrict__ x) {
  int idx = blockIdx.x * blockDim.x + threadIdx.x;
  if (idx >= kS * kB * kH) return;
  int h = idx % kH;
  int tmp = idx / kH;
  int b = tmp % kB;
  int t = tmp / kB;
  x[idx] = emb[(size_t)features[b * kS + t] * kH + h];
}

// One fused LSTM layer-step: z = in@Wx + hprev@Wh + bias ; cell update.
// Wave computes a 16x16 tile of each of the 4 gates (4 WMMA accumulators).
__global__ __launch_bounds__(128)
void lstm_step_kernel(const float* __restrict__ in,     // [B,H] layer input
                      const float* __restrict__ hprev,  // [B,H]
                      float* __restrict__ hnew,         // [B,H]
                      float* __restrict__ cst,          // [B,H] (in-place)
                      const float* __restrict__ W,      // [2H,4H]
                      const float* __restrict__ bias,   // [4H]
                      const int* __restrict__ seq_lengths,
                      float* __restrict__ outm,         // [B,S,H] or null
                      int t) {
  const int lane = threadIdx.x & 31;
  const int wave = threadIdx.x >> 5;
  const int lane16 = lane & 15;
  const int hi = lane >> 4;
  const int m0 = blockIdx.x * 16;
  const int n0 = (blockIdx.y * 4 + wave) * 16;
  const int ncol = n0 + lane16;

  v8f acc0 = {0, 0, 0, 0, 0, 0, 0, 0};
  v8f acc1 = acc0, acc2 = acc0, acc3 = acc0;

  const float* Arow = in + (size_t)(m0 + lane16) * kH + 2 * hi;
  const float* Hrow = hprev + (size_t)(m0 + lane16) * kH + 2 * hi;
  const float* W0 = W;                      // rows [0,H)   : input weights
  const float* W1 = W + (size_t)kH * k4H;   // rows [H,2H)  : hidden weights

  for (int k0 = 0; k0 < kH; k0 += 4) {
    v2f a = *(const v2f*)(Arow + k0);
    const float* Wk = W0 + (size_t)(k0 + 2 * hi) * k4H + ncol;
    v2f b0, b1, b2, b3;
    b0.x = Wk[0 * kH]; b0.y = Wk[k4H + 0 * kH];
    b1.x = Wk[1 * kH]; b1.y = Wk[k4H + 1 * kH];
    b2.x = Wk[2 * kH]; b2.y = Wk[k4H + 2 * kH];
    b3.x = Wk[3 * kH]; b3.y = Wk[k4H + 3 * kH];
    acc0 = wmma_f32(a, b0, acc0);
    acc1 = wmma_f32(a, b1, acc1);
    acc2 = wmma_f32(a, b2, acc2);
    acc3 = wmma_f32(a, b3, acc3);
  }
  for (int k0 = 0; k0 < kH; k0 += 4) {
    v2f a = *(const v2f*)(Hrow + k0);
    const float* Wk = W1 + (size_t)(k0 + 2 * hi) * k4H + ncol;
    v2f b0, b1, b2, b3;
    b0.x = Wk[0 * kH]; b0.y = Wk[k4H + 0 * kH];
    b1.x = Wk[1 * kH]; b1.y = Wk[k4H + 1 * kH];
    b2.x = Wk[2 * kH]; b2.y = Wk[k4H + 2 * kH];
    b3.x = Wk[3 * kH]; b3.y = Wk[k4H + 3 * kH];
    acc0 = wmma_f32(a, b0, acc0);
    acc1 = wmma_f32(a, b1, acc1);
    acc2 = wmma_f32(a, b2, acc2);
    acc3 = wmma_f32(a, b3, acc3);
  }

  const float bi = bias[0 * kH + ncol];
  const float bg = bias[1 * kH + ncol];
  const float bf = bias[2 * kH + ncol];
  const float bo = bias[3 * kH + ncol];

#pragma unroll
  for (int v = 0; v < 8; ++v) {
    int brow = m0 + v + 8 * hi;  // batch row for this accumulator element
    float zi = acc0[v] + bi;
    float zg = acc1[v] + bg;
    float zf = acc2[v] + bf;
    float zo = acc3[v] + bo;
    size_t idx = (size_t)brow * kH + ncol;
    float c_old = cst[idx];
    float cn = sigmoidf_(zf) * c_old + sigmoidf_(zi) * tanhf(zg);
    float hn = sigmoidf_(zo) * tanhf(cn);
    bool valid = t < seq_lengths[brow];
    cst[idx] = valid ? cn : c_old;
    hnew[idx] = valid ? hn : hprev[idx];
    if (outm) outm[((size_t)brow * kS + t) * kH + ncol] = valid ? hn : 0.0f;
  }
}

// Projection GEMM fused with online softmax partials + label-logit capture.
// grid = (160 row-tiles, 8); 8 waves/block => 64 column workers per row-tile.
__global__ __launch_bounds__(256)
void proj_softmax_kernel(const float* __restrict__ outm,  // [2560, 512]
                         const float* __restrict__ Wsm,   // [512, V]
                         const float* __restrict__ bsm,   // [V]
                         const int* __restrict__ labels,  // [2560] (b*S+s)
                         float* __restrict__ pm,          // [2560, 64]
                         float* __restrict__ ps,          // [2560, 64]
                         float* __restrict__ lab_logit) { // [2560]
  const int lane = threadIdx.x & 31;
  const int wave = threadIdx.x >> 5;
  const int lane16 = lane & 15;
  const int hi = lane >> 4;
  const int row0 = blockIdx.x * 16;
  const int worker = blockIdx.y * 8 + wave;

  int lbl[8];
#pragma unroll
  for (int v = 0; v < 8; ++v) lbl[v] = labels[row0 + v + 8 * hi];

  float rmax[8], rsum[8];
#pragma unroll
  for (int v = 0; v < 8; ++v) { rmax[v] = -__builtin_inff(); rsum[v] = 0.0f; }

  const float* Arow = outm + (size_t)(row0 + lane16) * kH + 2 * hi;

  for (int tIdx = worker; tIdx < kNTiles; tIdx += kWorkers) {
    const int ncol = tIdx * 16 + lane16;
    v8f acc = {0, 0, 0, 0, 0, 0, 0, 0};
    for (int k0 = 0; k0 < kH; k0 += 4) {
      v2f a = *(const v2f*)(Arow + k0);
      const float* Wk = Wsm + (size_t)(k0 + 2 * hi) * kV + ncol;
      v2f b;
      b.x = Wk[0];
      b.y = Wk[kV];
      acc = wmma_f32(a, b, acc);
    }
    float biasv = bsm[ncol];
#pragma unroll
    for (int v = 0; v < 8; ++v) {
      float val = acc[v] + biasv;
      if (ncol == lbl[v]) lab_logit[row0 + v + 8 * hi] = val;
      float mo = rmax[v];
      float mn = fmaxf(mo, val);
      rsum[v] = rsum[v] * __expf(mo - mn) + __expf(val - mn);
      rmax[v] = mn;
    }
  }

  // Reduce (max,sumexp) across the 16 lanes of each half (rows stay put).
#pragma unroll
  for (int off = 8; off >= 1; off >>= 1) {
#pragma unroll
    for (int v = 0; v < 8; ++v) {
      float om = __shfl_xor(rmax[v], off, 32);
      float os = __shfl_xor(rsum[v], off, 32);
      float mn = fmaxf(rmax[v], om);
      rsum[v] = rsum[v] * __expf(rmax[v] - mn) + os * __expf(om - mn);
      rmax[v] = mn;
    }
  }
  if (lane16 == 0) {
#pragma unroll
    for (int v = 0; v < 8; ++v) {
      int r = row0 + v + 8 * hi;
      pm[(size_t)r * kWorkers + worker] = rmax[v];
      ps[(size_t)r * kWorkers + worker] = rsum[v];
    }
  }
}

// Combine per-chunk (max,sumexp) partials into CE per token.
__global__ void ce_kernel(const float* __restrict__ pm, const float* __restrict__ ps,
                          const float* __restrict__ lab_logit, float* __restrict__ ce) {
  int r = blockIdx.x * blockDim.x + threadIdx.x;
  if (r >= kRows) return;
  float M = -__builtin_inff(), Ssum = 0.0f;
  for (int c = 0; c < kWorkers; ++c) {
    float m = pm[(size_t)r * kWorkers + c];
    float s = ps[(size_t)r * kWorkers + c];
    float mn = fmaxf(M, m);
    Ssum = Ssum * __expf(M - mn) + s * __expf(m - mn);
    M = mn;
  }
  ce[r] = M + __logf(Ssum) - lab_logit[r];
}

// d_out[0] = cost (mean over S of batch-averaged loss); d_out[1..128] = loss_list[B]
__global__ void finalize_kernel(const float* __restrict__ ce,
                                const float* __restrict__ mask,
                                float* __restrict__ out) {
  __shared__ float sh_loss[kS];
  int tid = threadIdx.x;
  if (tid < kB) {
    float num = 0.0f, den = 0.0f;
    for (int s = 0; s < kS; ++s) {
      float w = mask[tid * kS + s];
      num += ce[tid * kS + s] * w;
      den += w;
    }
    out[1 + tid] = num / den;
  }
  if (tid < kS) {
    float num = 0.0f, den = 0.0f;
    for (int b = 0; b < kB; ++b) {
      float w = mask[b * kS + tid];
      num += ce[b * kS + tid] * w;
      den += w;
    }
    sh_loss[tid] = num / den;
  }
  __syncthreads();
  if (tid == 0) {
    float cs = 0.0f;
    for (int s = 0; s < kS; ++s) cs += sh_loss[s];
    out[0] = cs / (float)kS;
  }
}

extern "C" void kernel_launch(void* const* d_in, const int* in_sizes, int n_in,
                              void* d_out, int out_size, void* d_ws, size_t ws_size,
                              hipStream_t stream) {
  const int* features = (const int*)d_in[0];      // [B,S]
  const int* labels = (const int*)d_in[1];        // [B,S]
  const int* seq_lengths = (const int*)d_in[2];   // [B]
  const float* mask = (const float*)d_in[3];      // [B,S]
  const float* embedding = (const float*)d_in[4]; // [V,H]
  const float* lstm_W = (const float*)d_in[5];    // [L,2H,4H]
  const float* lstm_b = (const float*)d_in[6];    // [L,4H]
  const float* softmax_w = (const float*)d_in[7]; // [H,V]
  const float* softmax_b = (const float*)d_in[8]; // [V]

  float* ws = (float*)d_ws;
  float* x = ws;                                  // [S,B,H]
  float* hbuf = x + (size_t)kS * kB * kH;         // [2][L][B][H] ping-pong
  float* cst = hbuf + (size_t)2 * kL * kB * kH;   // [L][B][H]
  float* outm = cst + (size_t)kL * kB * kH;       // [B,S,H]
  float* pm = outm + (size_t)kB * kS * kH;        // [2560,64]
  float* ps = pm + (size_t)kRows * kWorkers;      // [2560,64]
  float* lab = ps + (size_t)kRows * kWorkers;     // [2560]
  float* cebuf = lab + kRows;                     // [2560]

  // Deterministic: re-zero recurrent state every launch (ws is poisoned).
  int nz = 2 * kL * kB * kH + kL * kB * kH;  // hbuf + cst are contiguous
  zero_kernel<<<(nz + 255) / 256, 256, 0, stream>>>(hbuf, nz);

  int ne = kS * kB * kH;
  embed_kernel<<<(ne + 255) / 256, 256, 0, stream>>>(features, embedding, x);

  const size_t Wstride = (size_t)2 * kH * k4H;
  for (int t = 0; t < kS; ++t) {
    int pp = t & 1, cp = 1 - pp;
    // layer 0: input = x_t
    lstm_step_kernel<<<dim3(kB / 16, kH / 64), 128, 0, stream>>>(
        x + (size_t)t * kB * kH,
        hbuf + (size_t)(pp * kL + 0) * kB * kH,
        hbuf + (size_t)(cp * kL + 0) * kB * kH,
        cst,
        lstm_W, lstm_b, seq_lengths, nullptr, t);
    // layer 1: input = layer0's fresh h; writes masked output sequence
    lstm_step_kernel<<<dim3(kB / 16, kH / 64), 128, 0, stream>>>(
        hbuf + (size_t)(cp * kL + 0) * kB * kH,
        hbuf + (size_t)(pp * kL + 1) * kB * kH,
        hbuf + (size_t)(cp * kL + 1) * kB * kH,
        cst + (size_t)kB * kH,
        lstm_W + Wstride, lstm_b + k4H, seq_lengths, outm, t);
  }

  proj_softmax_kernel<<<dim3(kRowTiles, 8), 256, 0, stream>>>(
      outm, softmax_w, softmax_b, labels, pm, ps, lab);
  ce_kernel<<<(kRows + 255) / 256, 256, 0, stream>>>(pm, ps, lab, cebuf);
  finalize_kernel<<<1, 256, 0, stream>>>(cebuf, mask, (float*)d_out);
}